// V2VNetFusion_13932873908851
// MI455X (gfx1250) — compile-verified
//
#include <hip/hip_runtime.h>
#include <hip/hip_bf16.h>

typedef __attribute__((ext_vector_type(2))) float v2f;
typedef __attribute__((ext_vector_type(8))) float v8f;

#define B_  2
#define L_  5
#define C_  32
#define H_  48
#define W_  176
#define HW_ (H_ * W_)          // 8448 = 132 * 64
#define NTILE_ 132             // 64-pixel tiles per image

// ---------------------------------------------------------------------------
// Kernel 1: affine grid + bilinear warp + validity mask, only for ego i==0.
// warped[(b*L+j), c, h, w], mask[(b*L+j), h, w]
// ---------------------------------------------------------------------------
__global__ void warp_mask_kernel(const float* __restrict__ x,
                                 const float* __restrict__ Mmat,
                                 float* __restrict__ warped,
                                 float* __restrict__ mask)
{
    int idx = blockIdx.x * blockDim.x + threadIdx.x;   // over B*L*HW
    if (idx >= B_ * L_ * HW_) return;
    int p  = idx % HW_;
    int bj = idx / HW_;
    int j  = bj % L_;
    int b  = bj / L_;
    int h  = p / W_, w = p % W_;

    // M[b, 0, j] : (2,3)
    const float* M = Mmat + ((size_t)(b * L_) * L_ + j) * 6;
    float xs = (2.0f * (float)w + 1.0f) / (float)W_ - 1.0f;
    float ys = (2.0f * (float)h + 1.0f) / (float)H_ - 1.0f;
    float gx = M[0] * xs + M[1] * ys + M[2];
    float gy = M[3] * xs + M[4] * ys + M[5];
    float ix = ((gx + 1.0f) * (float)W_ - 1.0f) * 0.5f;
    float iy = ((gy + 1.0f) * (float)H_ - 1.0f) * 0.5f;
    float x0f = floorf(ix), y0f = floorf(iy);
    float wx = ix - x0f, wy = iy - y0f;
    int xi0 = (int)x0f, yi0 = (int)y0f;
    int xi1 = xi0 + 1,  yi1 = yi0 + 1;

    bool v00 = (xi0 >= 0 && xi0 < W_ && yi0 >= 0 && yi0 < H_);
    bool v10 = (xi1 >= 0 && xi1 < W_ && yi0 >= 0 && yi0 < H_);
    bool v01 = (xi0 >= 0 && xi0 < W_ && yi1 >= 0 && yi1 < H_);
    bool v11 = (xi1 >= 0 && xi1 < W_ && yi1 >= 0 && yi1 < H_);
    float w00 = (1.0f - wx) * (1.0f - wy), w10 = wx * (1.0f - wy);
    float w01 = (1.0f - wx) * wy,          w11 = wx * wy;

    mask[idx] = (v00 ? w00 : 0.0f) + (v10 ? w10 : 0.0f)
              + (v01 ? w01 : 0.0f) + (v11 ? w11 : 0.0f);

    int cx0 = xi0 < 0 ? 0 : (xi0 > W_ - 1 ? W_ - 1 : xi0);
    int cy0 = yi0 < 0 ? 0 : (yi0 > H_ - 1 ? H_ - 1 : yi0);
    int cx1 = xi1 < 0 ? 0 : (xi1 > W_ - 1 ? W_ - 1 : xi1);
    int cy1 = yi1 < 0 ? 0 : (yi1 > H_ - 1 ? H_ - 1 : yi1);
    float s00 = v00 ? w00 : 0.0f, s10 = v10 ? w10 : 0.0f;
    float s01 = v01 ? w01 : 0.0f, s11 = v11 ? w11 : 0.0f;

    const float* img = x + (size_t)(b * L_ + j) * C_ * HW_;   // feat[b, j]
    float* dst = warped + (size_t)bj * C_ * HW_ + p;
    int o00 = cy0 * W_ + cx0, o10 = cy0 * W_ + cx1;
    int o01 = cy1 * W_ + cx0, o11 = cy1 * W_ + cx1;
    for (int c = 0; c < C_; ++c) {
        const float* ic = img + (size_t)c * HW_;
        dst[(size_t)c * HW_] = ic[o00] * s00 + ic[o10] * s10
                             + ic[o01] * s01 + ic[o11] * s11;
    }
}

// ---------------------------------------------------------------------------
// Kernel 2: implicit-GEMM 3x3 conv (64->32 ch) via V_WMMA_F32_16X16X4_F32,
// fused mask-multiply, max over j, residual add. One wave = 16 pixels x 32 oc.
// K is iterated TAP-MAJOR (k' = tap*32 + channel within each 288-wide half),
// so tap offsets + zero-pad scales are hoisted to 9 per-lane registers and the
// inner loop is branch-free (load * scale). Ego K-half accumulated once.
// Weights are pre-permuted into tap-major order while staging into LDS.
// ---------------------------------------------------------------------------
__global__ void conv_fuse_kernel(const float* __restrict__ x,
                                 const float* __restrict__ warped,
                                 const float* __restrict__ mask,
                                 const float* __restrict__ msg_w,
                                 const float* __restrict__ msg_b,
                                 float* __restrict__ outfeat)
{
    __shared__ float wlds[32 * 288];   // 36 KB: one tap-major K-half

    int b    = blockIdx.x / NTILE_;
    int tile = blockIdx.x % NTILE_;
    int wave = threadIdx.x >> 5;       // wave32
    int lane = threadIdx.x & 31;
    int lane16 = lane & 15;
    int hi2    = (lane >> 4) << 1;     // K sub-offset: 0 (lanes 0-15) / 2 (16-31)
    int pbase  = tile * 64 + wave * 16;
    int pA     = pbase + lane16;       // A-matrix row = pixel
    int hA = pA / W_, wA = pA % W_;

    const float* ego = x + (size_t)(b * L_) * C_ * HW_;   // feat[b, 0]

    // per-tap clamped offsets + validity scales (fully unrolled -> registers)
    int   offs[9];
    float scal[9];
    #pragma unroll
    for (int r = 0; r < 9; ++r) {
        int dh = r / 3 - 1, dw = r % 3 - 1;               // compile-time
        int hh = hA + dh, ww = wA + dw;
        int hc = hh < 0 ? 0 : (hh > H_ - 1 ? H_ - 1 : hh);
        int wc = ww < 0 ? 0 : (ww > W_ - 1 ? W_ - 1 : ww);
        offs[r] = hc * W_ + wc;
        scal[r] = (hh == hc && ww == wc) ? 1.0f : 0.0f;
    }

    int nrow0 = lane16 * 288;           // B-matrix LDS row bases
    int nrow1 = (lane16 + 16) * 288;

    // ---- phase 0: stage ego-half weights tap-major: wlds[n*288+r*32+cl] ----
    for (int i = threadIdx.x; i < 32 * 288; i += blockDim.x) {
        int n = i / 288, t = i - n * 288;
        int r = t >> 5, cl = t & 31;
        wlds[i] = msg_w[n * 576 + (32 + cl) * 9 + r];
    }
    __syncthreads();

    // ---- ego contribution (shared across all j) ----
    v8f accE0 = {}, accE1 = {};
    #pragma unroll
    for (int r = 0; r < 9; ++r) {
        const float* base = ego + offs[r];
        float s = scal[r];
        #pragma unroll
        for (int cc = 0; cc < 32; cc += 4) {
            int c0 = cc + hi2;
            v2f a;  a.x = base[(size_t)c0 * HW_] * s;
                    a.y = base[(size_t)(c0 + 1) * HW_] * s;
            int kp = r * 32 + c0;
            v2f b0; b0.x = wlds[nrow0 + kp]; b0.y = wlds[nrow0 + kp + 1];
            v2f b1; b1.x = wlds[nrow1 + kp]; b1.y = wlds[nrow1 + kp + 1];
            accE0 = __builtin_amdgcn_wmma_f32_16x16x4_f32(false, a, false, b0,
                                                          (short)0, accE0, false, false);
            accE1 = __builtin_amdgcn_wmma_f32_16x16x4_f32(false, a, false, b1,
                                                          (short)0, accE1, false, false);
        }
    }
    __syncthreads();

    // ---- phase 1: stage warped-half weights tap-major ----
    for (int i = threadIdx.x; i < 32 * 288; i += blockDim.x) {
        int n = i / 288, t = i - n * 288;
        int r = t >> 5, cl = t & 31;
        wlds[i] = msg_w[n * 576 + cl * 9 + r];
    }
    __syncthreads();

    float bias0 = msg_b[lane16];
    float bias1 = msg_b[lane16 + 16];
    int   prow  = pbase + ((lane >> 4) << 3);   // C/D layout pixel base

    v8f best0, best1;
    #pragma unroll
    for (int v = 0; v < 8; ++v) { best0[v] = -3.402823466e38f; best1[v] = -3.402823466e38f; }

    for (int j = 0; j < L_; ++j) {
        const float* wimg = warped + (size_t)(b * L_ + j) * C_ * HW_;
        v8f acc0 = accE0, acc1 = accE1;
        #pragma unroll
        for (int r = 0; r < 9; ++r) {
            const float* base = wimg + offs[r];
            float s = scal[r];
            #pragma unroll
            for (int cc = 0; cc < 32; cc += 4) {
                int c0 = cc + hi2;
                v2f a;  a.x = base[(size_t)c0 * HW_] * s;
                        a.y = base[(size_t)(c0 + 1) * HW_] * s;
                int kp = r * 32 + c0;
                v2f b0; b0.x = wlds[nrow0 + kp]; b0.y = wlds[nrow0 + kp + 1];
                v2f b1; b1.x = wlds[nrow1 + kp]; b1.y = wlds[nrow1 + kp + 1];
                acc0 = __builtin_amdgcn_wmma_f32_16x16x4_f32(false, a, false, b0,
                                                             (short)0, acc0, false, false);
                acc1 = __builtin_amdgcn_wmma_f32_16x16x4_f32(false, a, false, b1,
                                                             (short)0, acc1, false, false);
            }
        }
        const float* mk = mask + (size_t)(b * L_ + j) * HW_;
        #pragma unroll
        for (int v = 0; v < 8; ++v) {
            float m  = mk[prow + v];
            float m0 = (acc0[v] + bias0) * m;
            float m1 = (acc1[v] + bias1) * m;
            best0[v] = fmaxf(best0[v], m0);
            best1[v] = fmaxf(best1[v], m1);
        }
    }

    // outfeat[b] = feat[b,0] + max_j(msg)
    float* of = outfeat + (size_t)b * C_ * HW_;
    #pragma unroll
    for (int v = 0; v < 8; ++v) {
        int p = prow + v;
        of[(size_t)lane16 * HW_ + p]        = ego[(size_t)lane16 * HW_ + p] + best0[v];
        of[(size_t)(lane16 + 16) * HW_ + p] = ego[(size_t)(lane16 + 16) * HW_ + p] + best1[v];
    }
}

// ---------------------------------------------------------------------------
// Kernel 3: 1x1 channel MLP out[b,d,p] = sum_c outfeat[b,c,p]*mlp_w[d,c] + mlp_b[d]
// Same WMMA scheme: M = 16 pixels, N = 32 out-ch, K = 32 in steps of 4.
// ---------------------------------------------------------------------------
__global__ void mlp_kernel(const float* __restrict__ outfeat,
                           const float* __restrict__ mlp_w,
                           const float* __restrict__ mlp_b,
                           float* __restrict__ out)
{
    int b    = blockIdx.x / NTILE_;
    int tile = blockIdx.x % NTILE_;
    int wave = threadIdx.x >> 5;
    int lane = threadIdx.x & 31;
    int lane16 = lane & 15;
    int hi2    = (lane >> 4) << 1;
    int pbase  = tile * 64 + wave * 16;
    int pA     = pbase + lane16;

    const float* fin = outfeat + (size_t)b * C_ * HW_;
    v8f acc0 = {}, acc1 = {};
    #pragma unroll
    for (int kk = 0; kk < 32; kk += 4) {
        int k0 = kk + hi2;
        v2f a;  a.x = fin[(size_t)k0 * HW_ + pA];
                a.y = fin[(size_t)(k0 + 1) * HW_ + pA];
        v2f b0; b0.x = mlp_w[lane16 * 32 + k0];
                b0.y = mlp_w[lane16 * 32 + k0 + 1];
        v2f b1; b1.x = mlp_w[(lane16 + 16) * 32 + k0];
                b1.y = mlp_w[(lane16 + 16) * 32 + k0 + 1];
        acc0 = __builtin_amdgcn_wmma_f32_16x16x4_f32(false, a, false, b0,
                                                     (short)0, acc0, false, false);
        acc1 = __builtin_amdgcn_wmma_f32_16x16x4_f32(false, a, false, b1,
                                                     (short)0, acc1, false, false);
    }

    int prow = pbase + ((lane >> 4) << 3);
    float bias0 = mlp_b[lane16], bias1 = mlp_b[lane16 + 16];
    float* o = out + (size_t)b * C_ * HW_;
    #pragma unroll
    for (int v = 0; v < 8; ++v) {
        o[(size_t)lane16 * HW_ + prow + v]        = acc0[v] + bias0;
        o[(size_t)(lane16 + 16) * HW_ + prow + v] = acc1[v] + bias1;
    }
}

extern "C" void kernel_launch(void* const* d_in, const int* in_sizes, int n_in,
                              void* d_out, int out_size, void* d_ws, size_t ws_size,
                              hipStream_t stream)
{
    (void)in_sizes; (void)n_in; (void)out_size; (void)ws_size;
    const float* x     = (const float*)d_in[0];
    // d_in[1] = record_len (int64) — always full L, unused by reference math
    const float* M     = (const float*)d_in[2];
    const float* msg_w = (const float*)d_in[3];
    const float* msg_b = (const float*)d_in[4];
    const float* mlp_w = (const float*)d_in[5];
    const float* mlp_b = (const float*)d_in[6];
    float* out = (float*)d_out;

    float* warped  = (float*)d_ws;                          // B*L*C*HW
    float* mask    = warped + (size_t)B_ * L_ * C_ * HW_;   // B*L*HW
    float* outfeat = mask   + (size_t)B_ * L_ * HW_;        // B*C*HW

    int n1 = B_ * L_ * HW_;
    warp_mask_kernel<<<(n1 + 255) / 256, 256, 0, stream>>>(x, M, warped, mask);
    conv_fuse_kernel<<<B_ * NTILE_, 128, 0, stream>>>(x, warped, mask, msg_w, msg_b, outfeat);
    mlp_kernel<<<B_ * NTILE_, 128, 0, stream>>>(outfeat, mlp_w, mlp_b, out);
}